// RNN_5_26499948216344
// MI455X (gfx1250) — compile-verified
//
#include <hip/hip_runtime.h>
#include <hip/hip_bf16.h>

typedef __attribute__((ext_vector_type(2))) float v2f;
typedef __attribute__((ext_vector_type(8))) float v8f;

#define SEQ     524288
#define IN_DIM  5
#define HID     10

__device__ __forceinline__ float gpu_tanh(float x) {
#if __has_builtin(__builtin_amdgcn_tanhf)
  return __builtin_amdgcn_tanhf(x);   // v_tanh_f32 (gfx1250 trans op)
#else
  return tanhf(x);
#endif
}

__device__ __forceinline__ float bcast_lane(float v, int k) {
  return __uint_as_float(__builtin_amdgcn_readlane(__float_as_uint(v), k));
}

// ---------------------------------------------------------------------------
// Kernel 1: xp[t][n] = sum_k src[t][k] * W_ih[n][k] + b_ih[n] + b_hh[n]
// One wave handles 16 timesteps via V_WMMA_F32_16X16X4_F32 (two WMMAs: K=0..3
// and K=4 zero-padded).  A is 16x4 f32: lanes 0-15 hold (K0,K1), lanes 16-31
// hold (K2,K3).  B is 4x16: VGPR0 = rows {0,2}, VGPR1 = rows {1,3} striped
// across lane halves.  C/D: VGPR r -> row M = r + 8*(lane/16), col N = lane%16.
// ---------------------------------------------------------------------------
__global__ __launch_bounds__(256) void rnn_inproj_wmma(
    const float* __restrict__ src, const float* __restrict__ W_ih,
    const float* __restrict__ b_ih, const float* __restrict__ b_hh,
    float* __restrict__ xp) {
  const int lane = threadIdx.x & 31;
  const int wave = (blockIdx.x * blockDim.x + threadIdx.x) >> 5;
  const int t0   = wave * 16;
  const int half = lane >> 4;   // 0: K rows 0/1, 1: K rows 2/3
  const int l    = lane & 15;   // A: row M, B: col N

  // A matrices (timestep rows of src)
  const float* s = src + (size_t)(t0 + l) * IN_DIM;
  v2f a1; a1.x = s[half ? 2 : 0]; a1.y = s[half ? 3 : 1];
  v2f a2; a2.x = half ? 0.f : s[4]; a2.y = 0.f;          // K=4 in slot K0

  // B matrices: B[k][n] = W_ih[n][k], zero for n >= HID
  v2f b1, b2;
  b1.x = b1.y = b2.x = b2.y = 0.f;
  if (l < HID) {
    const float* w = W_ih + l * IN_DIM;
    b1.x = w[half ? 2 : 0];
    b1.y = w[half ? 3 : 1];
    b2.x = half ? 0.f : w[4];                            // row K0 of 1st (K=4) WMMA
  }

  v8f c = {};
  c = __builtin_amdgcn_wmma_f32_16x16x4_f32(false, a2, false, b2,
                                            (short)0, c, false, false);
  c = __builtin_amdgcn_wmma_f32_16x16x4_f32(false, a1, false, b1,
                                            (short)0, c, false, false);

  if (l < HID) {
    const float bias = b_ih[l] + b_hh[l];
#pragma unroll
    for (int r = 0; r < 8; ++r) {
      const int t = t0 + r + 8 * half;
      xp[(size_t)t * HID + l] = c[r] + bias;
    }
  }
}

// ---------------------------------------------------------------------------
// Kernel 2: the serial scan.  One wave32.  Lane j (j<10) owns row j of W_hh in
// VGPRs; h lives as uniform (SGPR) values refreshed by v_readlane each step.
//
// Latency-critical per-step chain: tanh -> readlane -> dot(W_hh[j,:], h).
// The 10-term dot is split into 3 partial accumulators (depth 4 FMA + 2 adds
// instead of 10 dependent FMAs) to cut the serial dependency latency ~2x.
// The output projection (all-uniform math) is off the h critical path; its
// scalar result is slotted into lane (t%32) and stored coalesced every 32
// steps.  Unroll-by-32 lets the compiler batch the independent xp loads.
// ---------------------------------------------------------------------------
__global__ __launch_bounds__(32) void rnn_scan(
    const float* __restrict__ xp, const float* __restrict__ W_hh,
    const float* __restrict__ W_fc, const float* __restrict__ b_fc,
    float* __restrict__ out) {
  const int lane = threadIdx.x;

  float w[HID];
#pragma unroll
  for (int k = 0; k < HID; ++k)
    w[k] = (lane < HID) ? W_hh[lane * HID + k] : 0.f;

  float wfc[HID], h[HID];
#pragma unroll
  for (int k = 0; k < HID; ++k) { wfc[k] = W_fc[k]; h[k] = 0.f; }
  const float bfc = b_fc[0];

  const int xidx = (lane < HID) ? lane : 0;   // clamp lanes 10..31 in-range
  float obuf = 0.f;

  for (int t0 = 0; t0 < SEQ; t0 += 32) {
#pragma unroll
    for (int u = 0; u < 32; ++u) {
      const int t = t0 + u;
      const float xpv = xp[(size_t)t * HID + xidx];   // lane j: xp[t][j]

      // 3-way split accumulation: depth = 4 FMA + 2 ADD on the serial chain.
      float p0 = xpv;
      p0 = fmaf(h[0], w[0], p0);
      p0 = fmaf(h[3], w[3], p0);
      p0 = fmaf(h[6], w[6], p0);
      p0 = fmaf(h[9], w[9], p0);
      float p1 = h[1] * w[1];
      p1 = fmaf(h[4], w[4], p1);
      p1 = fmaf(h[7], w[7], p1);
      float p2 = h[2] * w[2];
      p2 = fmaf(h[5], w[5], p2);
      p2 = fmaf(h[8], w[8], p2);
      const float acc = (p0 + p1) + p2;

      const float hn = gpu_tanh(acc);

      // h -> uniform SGPRs; h[0..2] first so next step's partial chains
      // can start as early as possible.
      h[0] = bcast_lane(hn, 0);
      h[1] = bcast_lane(hn, 1);
      h[2] = bcast_lane(hn, 2);
      h[3] = bcast_lane(hn, 3);
      h[4] = bcast_lane(hn, 4);
      h[5] = bcast_lane(hn, 5);
      h[6] = bcast_lane(hn, 6);
      h[7] = bcast_lane(hn, 7);
      h[8] = bcast_lane(hn, 8);
      h[9] = bcast_lane(hn, 9);

      // out[t] = W_fc . h_t + b_fc  (uniform math, off the critical path)
      float o0 = fmaf(h[0], wfc[0], bfc);
      o0 = fmaf(h[2], wfc[2], o0);
      o0 = fmaf(h[4], wfc[4], o0);
      o0 = fmaf(h[6], wfc[6], o0);
      o0 = fmaf(h[8], wfc[8], o0);
      float o1 = h[1] * wfc[1];
      o1 = fmaf(h[3], wfc[3], o1);
      o1 = fmaf(h[5], wfc[5], o1);
      o1 = fmaf(h[7], wfc[7], o1);
      o1 = fmaf(h[9], wfc[9], o1);
      const float o = o0 + o1;

      obuf = (lane == u) ? o : obuf;                 // slot into lane u
    }
    out[t0 + lane] = obuf;                           // coalesced 128B store
  }
}

// ---------------------------------------------------------------------------
extern "C" void kernel_launch(void* const* d_in, const int* in_sizes, int n_in,
                              void* d_out, int out_size, void* d_ws,
                              size_t ws_size, hipStream_t stream) {
  const float* src  = (const float*)d_in[0];   // (SEQ,1,5)
  const float* W_ih = (const float*)d_in[1];   // (10,5)
  const float* W_hh = (const float*)d_in[2];   // (10,10)
  const float* b_ih = (const float*)d_in[3];   // (10,)
  const float* b_hh = (const float*)d_in[4];   // (10,)
  const float* W_fc = (const float*)d_in[5];   // (1,10)
  const float* b_fc = (const float*)d_in[6];   // (1,)
  float*       out  = (float*)d_out;           // SEQ floats
  float*       xp   = (float*)d_ws;            // SEQ*HID floats = ~21 MB scratch

  // 16 timesteps per wave, 8 waves per 256-thread block.
  const int waves  = SEQ / 16;
  const int blocks = waves / 8;
  rnn_inproj_wmma<<<blocks, 256, 0, stream>>>(src, W_ih, b_ih, b_hh, xp);
  rnn_scan<<<1, 32, 0, stream>>>(xp, W_hh, W_fc, b_fc, out);
}